// ConvLayer_39402029973527
// MI455X (gfx1250) — compile-verified
//
#include <hip/hip_runtime.h>

#define NN 8192
#define C  32
#define KSPLIT 4                 // K-dimension split across blocks
#define KCHUNK (NN / KSPLIT)     // 2048
#define ITERS  (KCHUNK / 32)     // 64 iterations per wave per matrix (8 waves x K=4)

typedef float v2f __attribute__((ext_vector_type(2)));
typedef float v8f __attribute__((ext_vector_type(8)));

// ---------------------------------------------------------------------------
// Kernel 1: Yt[k][c][n] = sum_i X[n][i] * Wk[i][c]   (transposed small GEMM)
// ---------------------------------------------------------------------------
__global__ __launch_bounds__(256) void small_gemm_kernel(
    const float* __restrict__ X,
    const float* __restrict__ W0,
    const float* __restrict__ W1,
    const float* __restrict__ W2,
    float* __restrict__ Yt)
{
    const int gid = blockIdx.x * 256 + threadIdx.x;
    const int n = gid & (NN - 1);
    const int c = (gid >> 13) & 31;
    const int k = gid >> 18;

    const float* __restrict__ W = (k == 0) ? W0 : ((k == 1) ? W1 : W2);
    const float* __restrict__ xr = X + (size_t)n * C;

    float s = 0.0f;
#pragma unroll
    for (int i = 0; i < C; ++i)
        s = fmaf(xr[i], W[i * C + c], s);

    Yt[gid] = s;
}

// ---------------------------------------------------------------------------
// Kernel 2 (pass 1): partial[kslice][m][c] = sum over K-slice of Lk rows x Yt
// grid = 512 row-blocks x KSPLIT.  8 waves interleave K with stride 4.
// Uniform compile-time trip count -> real unrolling -> deep load pipelining.
// ---------------------------------------------------------------------------
__global__ __launch_bounds__(256) void agg_pass1_kernel(
    const float* __restrict__ L0,
    const float* __restrict__ L1,
    const float* __restrict__ L2,
    const float* __restrict__ Yt,      // [3][32][8192]
    float* __restrict__ partial)       // [KSPLIT][8192][32]
{
    __shared__ float red[2 * 8 * 512];   // 32 KB

    const int tid    = threadIdx.x;
    const int wave   = tid >> 5;
    const int lane   = tid & 31;
    const int rowblk = blockIdx.x >> 2;          // / KSPLIT
    const int kslice = blockIdx.x & (KSPLIT - 1);
    const int row0   = rowblk * 16;
    const int kbase  = kslice * KCHUNK + wave * 4;   // wave-uniform start

    // A-operand lane mapping for 16x4 f32 tiles:
    //   lanes 0-15  : row M=lane,    K = +0 (v0), +1 (v1)
    //   lanes 16-31 : row M=lane-16, K = +2 (v0), +3 (v1)
    const int lrow = lane & 15;
    const int lk   = (lane >> 4) * 2;

    v8f acc0 = {};   // cols 0..15
    v8f acc1 = {};   // cols 16..31

    const float* Ls[3] = { L0, L1, L2 };

    for (int k = 0; k < 3; ++k) {
        const float* __restrict__ pA  = Ls[k] + (size_t)(row0 + lrow) * NN + lk + kbase;
        const float* __restrict__ pB0 = Yt + (size_t)k * C * NN + (size_t)lrow * NN + lk + kbase;
        const float* __restrict__ pB1 = pB0 + (size_t)16 * NN;

#pragma unroll 16
        for (int it = 0; it < ITERS; ++it) {
            v2f a  = __builtin_nontemporal_load((const v2f*)pA);   // L: stream once
            v2f b0 = *(const v2f*)pB0;                             // Yt: L2-resident
            v2f b1 = *(const v2f*)pB1;
            pA += 32; pB0 += 32; pB1 += 32;
            acc0 = __builtin_amdgcn_wmma_f32_16x16x4_f32(
                       false, a, false, b0, (short)0, acc0, false, false);
            acc1 = __builtin_amdgcn_wmma_f32_16x16x4_f32(
                       false, a, false, b1, (short)0, acc1, false, false);
        }
    }

    // spill partial 16x16 tiles: red[half*4096 + wave*512 + lane*8 + g]
#pragma unroll
    for (int i = 0; i < 8; ++i) {
        red[wave * 512 + lane * 8 + i]        = acc0[i];
        red[4096 + wave * 512 + lane * 8 + i] = acc1[i];
    }
    __syncthreads();

    // reduce the 8 waves; store raw partial tile (bias/relu applied in pass 2)
    // C/D layout: VGPR g, lane l<16 -> M=g, N=l ; lane>=16 -> M=g+8, N=l-16
#pragma unroll
    for (int e = tid; e < 512; e += 256) {
        const int m    = e >> 5;
        const int c    = e & 31;
        const int half = c >> 4;
        const int g    = m & 7;
        const int rl   = ((m >> 3) << 4) | (c & 15);
        float s = 0.0f;
#pragma unroll
        for (int w = 0; w < 8; ++w)
            s += red[half * 4096 + w * 512 + rl * 8 + g];
        partial[(size_t)kslice * NN * C + (size_t)(row0 + m) * C + c] = s;
    }
}

// ---------------------------------------------------------------------------
// Kernel 3: out = relu(sum over KSPLIT partials + bias)
// ---------------------------------------------------------------------------
__global__ __launch_bounds__(256) void reduce_bias_relu_kernel(
    const float* __restrict__ partial,
    const float* __restrict__ bias,
    float* __restrict__ out)
{
    const int e = blockIdx.x * 256 + threadIdx.x;   // 0 .. 8192*32-1
    float s = bias[e & 31];
#pragma unroll
    for (int p = 0; p < KSPLIT; ++p)
        s += partial[(size_t)p * NN * C + e];
    out[e] = fmaxf(s, 0.0f);
}

// ---------------------------------------------------------------------------
extern "C" void kernel_launch(void* const* d_in, const int* in_sizes, int n_in,
                              void* d_out, int out_size, void* d_ws, size_t ws_size,
                              hipStream_t stream)
{
    const float* X    = (const float*)d_in[0];
    const float* L0   = (const float*)d_in[1];
    const float* L1   = (const float*)d_in[2];
    const float* L2   = (const float*)d_in[3];
    const float* W0   = (const float*)d_in[4];
    const float* W1   = (const float*)d_in[5];
    const float* W2   = (const float*)d_in[6];
    const float* bias = (const float*)d_in[7];

    float* Yt      = (float*)d_ws;                 // 3*32*8192 floats = 3 MB
    float* partial = Yt + (size_t)3 * C * NN;      // KSPLIT*8192*32 floats = 4 MB

    small_gemm_kernel<<<(3 * C * NN) / 256, 256, 0, stream>>>(X, W0, W1, W2, Yt);
    agg_pass1_kernel<<<(NN / 16) * KSPLIT, 256, 0, stream>>>(L0, L1, L2, Yt, partial);
    reduce_bias_relu_kernel<<<(NN * C) / 256, 256, 0, stream>>>(partial, bias, (float*)d_out);
}